// Gating_10952166605135
// MI455X (gfx1250) — compile-verified
//
#include <hip/hip_runtime.h>
#include <hip/hip_bf16.h>

typedef __attribute__((ext_vector_type(2))) float v2f;
typedef __attribute__((ext_vector_type(8))) float v8f;

// ---------------------------------------------------------------------------
// Degree accumulation:  deg[dst[e]] += 1
// ---------------------------------------------------------------------------
__global__ void deg_kernel(const int* __restrict__ dst, float* __restrict__ deg, int E) {
    int e = blockIdx.x * blockDim.x + threadIdx.x;
    if (e < E) atomicAdd(&deg[dst[e]], 1.0f);
}

// deg -> dinv = rsqrt(deg + 1)   (self-loop degree)
__global__ void rsqrt_kernel(float* __restrict__ d, int N) {
    int i = blockIdx.x * blockDim.x + threadIdx.x;
    if (i < N) d[i] = rsqrtf(d[i] + 1.0f);
}

// ---------------------------------------------------------------------------
// WMMA f32 GEMM:  C[M x NCOLS] = A[M x K] * W[K x NCOLS]   (fp32-exact path)
// One wave per 16-row strip; NT = NCOLS/16 accumulator tiles; K stepped by 4
// using V_WMMA_F32_16X16X4_F32.
// A fragment layout (16x4 f32, ISA 7.12.2): lane<16 -> M=lane, K=v+0..1;
// lane>=16 -> M=lane-16, K=v+2..3.  B (4x16) mirrors with M->N.
// ---------------------------------------------------------------------------
template <int K, int NCOLS>
__global__ void wmma_gemm_f32(const float* __restrict__ A,
                              const float* __restrict__ W,
                              float* __restrict__ C, int M) {
    constexpr int NT = NCOLS / 16;
    const int wavesPerBlock = blockDim.x >> 5;
    const int wid  = blockIdx.x * wavesPerBlock + (threadIdx.x >> 5);
    const int lane = threadIdx.x & 31;
    const int m0   = wid * 16;
    if (m0 >= M) return;                 // wave-uniform exit: EXEC stays all-1s

    const int half = lane >> 4;          // 0: K pair {0,1} / rows 0..7; 1: {2,3} / rows 8..15
    const int idx  = lane & 15;

    v8f acc[NT];
#pragma unroll
    for (int t = 0; t < NT; ++t) acc[t] = (v8f)0.0f;

    const float* arow = A + (size_t)(m0 + idx) * K + 2 * half;

    for (int k0 = 0; k0 < K; k0 += 4) {
        v2f a;
        a.x = arow[k0 + 0];
        a.y = arow[k0 + 1];
        const int krow = k0 + 2 * half;
#pragma unroll
        for (int t = 0; t < NT; ++t) {
            v2f b;
            b.x = W[(size_t)krow * NCOLS + t * 16 + idx];
            b.y = W[(size_t)(krow + 1) * NCOLS + t * 16 + idx];
            acc[t] = __builtin_amdgcn_wmma_f32_16x16x4_f32(
                false, a, false, b, (short)0, acc[t], false, false);
        }
    }

#pragma unroll
    for (int t = 0; t < NT; ++t) {
#pragma unroll
        for (int r = 0; r < 8; ++r) {
            C[(size_t)(m0 + r + 8 * half) * NCOLS + t * 16 + idx] = acc[t][r];
        }
    }
}

// ---------------------------------------------------------------------------
// h[i,c] = dinv[i]^2 * xw[i,c] + bias[c]   (self-loop message + bias)
// ---------------------------------------------------------------------------
template <int C>
__global__ void selfloop_init(const float* __restrict__ xw,
                              const float* __restrict__ dinv,
                              const float* __restrict__ bias,
                              float* __restrict__ h, int N) {
    long long i = (long long)blockIdx.x * blockDim.x + threadIdx.x;
    long long total = (long long)N * C;
    if (i >= total) return;
    int node = (int)(i / C);
    int c    = (int)(i % C);
    float di = dinv[node];
    h[i] = di * di * xw[i] + bias[c];
}

// ---------------------------------------------------------------------------
// h[dst,c] += xw[src,c] * dinv[src]*dinv[dst]   over all edges
// ---------------------------------------------------------------------------
template <int C>
__global__ void edge_scatter(const int* __restrict__ src,
                             const int* __restrict__ dst,
                             const float* __restrict__ xw,
                             const float* __restrict__ dinv,
                             float* __restrict__ h, int E) {
    long long i = (long long)blockIdx.x * blockDim.x + threadIdx.x;
    long long total = (long long)E * C;
    if (i >= total) return;
    int e = (int)(i / C);
    int c = (int)(i % C);
    int s = src[e], d = dst[e];
    float nrm = dinv[s] * dinv[d];
    atomicAdd(&h[(size_t)d * C + c], xw[(size_t)s * C + c] * nrm);
}

// ---------------------------------------------------------------------------
// Mean-pool numerator + counts: pooled[g, colOff+c] += h2[i,c]; cnt[g] += 1
// ---------------------------------------------------------------------------
__global__ void pool_kernel(const float* __restrict__ h2,
                            const int* __restrict__ batch,
                            float* __restrict__ pooled,
                            float* __restrict__ cnt, int N, int colOff) {
    int i = blockIdx.x * blockDim.x + threadIdx.x;
    int node = i >> 5;        // 32 channels per node
    int c    = i & 31;
    if (node >= N) return;
    int g = batch[node];
    atomicAdd(&pooled[(size_t)g * 64 + colOff + c], h2[(size_t)node * 32 + c]);
    if (c == 0) atomicAdd(&cnt[g], 1.0f);
}

// ---------------------------------------------------------------------------
// gate = softmax( (pooled/cnt) @ Wc + bc )   — one thread per graph (B=4096)
// ---------------------------------------------------------------------------
__global__ void gate_kernel(const float* __restrict__ pooled,
                            const float* __restrict__ cnt0,
                            const float* __restrict__ cnt1,
                            const float* __restrict__ Wc,
                            const float* __restrict__ bc,
                            float* __restrict__ gate, int B) {
    int g = blockIdx.x * blockDim.x + threadIdx.x;
    if (g >= B) return;
    float inv0 = 1.0f / fmaxf(cnt0[g], 1.0f);
    float inv1 = 1.0f / fmaxf(cnt1[g], 1.0f);
    float logits[8];
#pragma unroll
    for (int e = 0; e < 8; ++e) logits[e] = bc[e];
    const float* row = pooled + (size_t)g * 64;
    for (int c = 0; c < 64; ++c) {
        float x = row[c] * (c < 32 ? inv0 : inv1);
#pragma unroll
        for (int e = 0; e < 8; ++e) logits[e] += x * Wc[c * 8 + e];
    }
    float m = logits[0];
#pragma unroll
    for (int e = 1; e < 8; ++e) m = fmaxf(m, logits[e]);
    float ex[8];
    float sum = 0.0f;
#pragma unroll
    for (int e = 0; e < 8; ++e) { ex[e] = __expf(logits[e] - m); sum += ex[e]; }
    float inv = 1.0f / sum;
#pragma unroll
    for (int e = 0; e < 8; ++e) gate[(size_t)g * 8 + e] = ex[e] * inv;
}

// ---------------------------------------------------------------------------
// Distortion loss: one wave32 per graph-edge.
// Lane l covers embedding dims [16l, 16l+16) -> expert e = l>>2 (4 lanes/expert).
// Group-of-4 shuffle reduce -> per-expert sq distance; 8-wide in-register
// softmax of gate[s]*gate[d]; |sum(diff*w)/dis - 1| atomically accumulated.
// ---------------------------------------------------------------------------
__global__ void distortion_kernel(const float* __restrict__ emb,
                                  const float* __restrict__ gate,
                                  const int* __restrict__ sidx,
                                  const int* __restrict__ didx,
                                  const float* __restrict__ dis,
                                  float* __restrict__ lossAcc, int E2) {
    int wid  = (int)(((long long)blockIdx.x * blockDim.x + threadIdx.x) >> 5);
    int lane = threadIdx.x & 31;
    if (wid >= E2) return;                // wave-uniform exit

    int s = sidx[wid], d = didx[wid];
    const float* es = emb + (size_t)s * 512;
    const float* ed = emb + (size_t)d * 512;
    __builtin_prefetch(es + lane * 16, 0, 1);
    __builtin_prefetch(ed + lane * 16, 0, 1);

    float part = 0.0f;
    const int base = lane * 16;
#pragma unroll
    for (int j = 0; j < 16; ++j) {
        float df = es[base + j] - ed[base + j];
        part += df * df;
    }
    // reduce within each group of 4 lanes (one expert per group)
    part += __shfl_xor(part, 1, 32);
    part += __shfl_xor(part, 2, 32);
    // lane l fetches expert (l&7)'s sum from lane 4*(l&7)
    float diff_e = __shfl(part, (lane & 7) * 4, 32);

    int e = lane & 7;
    float p = gate[(size_t)s * 8 + e] * gate[(size_t)d * 8 + e];
    // 8-wide softmax (replicated across the 4 groups of 8 lanes)
    float m = p;
    m = fmaxf(m, __shfl_xor(m, 1, 32));
    m = fmaxf(m, __shfl_xor(m, 2, 32));
    m = fmaxf(m, __shfl_xor(m, 4, 32));
    float ex = __expf(p - m);
    float sum = ex;
    sum += __shfl_xor(sum, 1, 32);
    sum += __shfl_xor(sum, 2, 32);
    sum += __shfl_xor(sum, 4, 32);
    float contrib = diff_e * (ex / sum);
    contrib += __shfl_xor(contrib, 1, 32);
    contrib += __shfl_xor(contrib, 2, 32);
    contrib += __shfl_xor(contrib, 4, 32);

    if (lane == 0) {
        atomicAdd(lossAcc, fabsf(contrib / dis[wid] - 1.0f));
    }
}

__global__ void finalize_loss(const float* __restrict__ lossAcc,
                              float* __restrict__ out, float invE2) {
    if (blockIdx.x == 0 && threadIdx.x == 0) out[0] = lossAcc[0] * invE2;
}

// ---------------------------------------------------------------------------
// Orchestration
// ---------------------------------------------------------------------------
extern "C" void kernel_launch(void* const* d_in, const int* in_sizes, int n_in,
                              void* d_out, int out_size, void* d_ws, size_t ws_size,
                              hipStream_t stream) {
    const float* x0  = (const float*)d_in[0];
    const float* x1  = (const float*)d_in[1];
    const float* W1  = (const float*)d_in[2];
    const float* b1  = (const float*)d_in[3];
    const float* W2  = (const float*)d_in[4];
    const float* b2  = (const float*)d_in[5];
    const float* Wc  = (const float*)d_in[6];
    const float* bc  = (const float*)d_in[7];
    const float* emb = (const float*)d_in[8];
    const float* dis = (const float*)d_in[9];
    const int* ei0   = (const int*)d_in[10];
    const int* ei1   = (const int*)d_in[11];
    const int* bat0  = (const int*)d_in[12];
    const int* bat1  = (const int*)d_in[13];
    const int* ei2   = (const int*)d_in[14];

    const int N  = in_sizes[0] / 64;   // 50000 (divisible by 16)
    const int E  = in_sizes[10] / 2;   // 800000
    const int E2 = in_sizes[9];        // 131072
    const int B  = in_sizes[8] / 512;  // 4096

    // workspace layout (floats)
    float* ws      = (float*)d_ws;
    float* dinv    = ws;                               // N
    float* bufXW   = dinv + N;                         // N*64 (xw1; later xw2 | h2)
    float* bufH1   = bufXW + (size_t)N * 64;           // N*64
    float* pooled  = bufH1 + (size_t)N * 64;           // B*64
    float* cnt0    = pooled + (size_t)B * 64;          // B
    float* cnt1    = cnt0 + B;                         // B
    float* lossAcc = cnt1 + B;                         // 1

    float* gate    = (float*)d_out;                    // B*8
    float* lossOut = gate + (size_t)B * 8;             // 1

    // zero pooled + cnt0 + cnt1 + lossAcc (contiguous)
    hipMemsetAsync(pooled, 0, (size_t)(B * 64 + 2 * B + 1) * sizeof(float), stream);

    const int tilesM    = N / 16;
    const int gemmGrid  = (tilesM + 7) / 8;            // 8 waves (256 threads) / block

    for (int v = 0; v < 2; ++v) {
        const float* x    = v ? x1 : x0;
        const int* src    = v ? ei1 : ei0;
        const int* dstp   = src + E;
        const int* batch  = v ? bat1 : bat0;
        float* cnt        = v ? cnt1 : cnt0;

        // degree -> dinv
        hipMemsetAsync(dinv, 0, (size_t)N * sizeof(float), stream);
        deg_kernel<<<(E + 255) / 256, 256, 0, stream>>>(dstp, dinv, E);
        rsqrt_kernel<<<(N + 255) / 256, 256, 0, stream>>>(dinv, N);

        // conv1: xw1 = x @ W1 ; h1 = scatter + self + b1
        wmma_gemm_f32<64, 64><<<gemmGrid, 256, 0, stream>>>(x, W1, bufXW, N);
        {
            long long tot = (long long)N * 64;
            selfloop_init<64><<<(int)((tot + 255) / 256), 256, 0, stream>>>(bufXW, dinv, b1, bufH1, N);
            long long etot = (long long)E * 64;
            edge_scatter<64><<<(int)((etot + 255) / 256), 256, 0, stream>>>(src, dstp, bufXW, dinv, bufH1, E);
        }

        // conv2: xw2 = h1 @ W2 ; h2 = scatter + self + b2
        float* xw2 = bufXW;                    // N*32
        float* h2  = bufXW + (size_t)N * 32;   // N*32
        wmma_gemm_f32<64, 32><<<gemmGrid, 256, 0, stream>>>(bufH1, W2, xw2, N);
        {
            long long tot = (long long)N * 32;
            selfloop_init<32><<<(int)((tot + 255) / 256), 256, 0, stream>>>(xw2, dinv, b2, h2, N);
            long long etot = (long long)E * 32;
            edge_scatter<32><<<(int)((etot + 255) / 256), 256, 0, stream>>>(src, dstp, xw2, dinv, h2, E);
        }

        // mean-pool numerator + counts
        {
            long long tot = (long long)N * 32;
            pool_kernel<<<(int)((tot + 255) / 256), 256, 0, stream>>>(h2, batch, pooled, cnt, N, v * 32);
        }
    }

    // gate = softmax(pooled_mean @ Wc + bc)  -> d_out[0 .. B*8)
    gate_kernel<<<(B + 255) / 256, 256, 0, stream>>>(pooled, cnt0, cnt1, Wc, bc, gate, B);

    // distortion loss over E2 edges: one wave per edge
    {
        long long tot = (long long)E2 * 32;
        distortion_kernel<<<(int)((tot + 255) / 256), 256, 0, stream>>>(
            emb, gate, ei2, ei2 + E2, dis, lossAcc, E2);
    }
    finalize_loss<<<1, 32, 0, stream>>>(lossAcc, lossOut, 1.0f / (float)E2);
}